// ActorCriticGNN_80796924772854
// MI455X (gfx1250) — compile-verified
//
#include <hip/hip_runtime.h>
#include <stdint.h>

// Problem constants (from the reference)
#define N_NODES 50000
#define DIM     256
#define N_EDGES 800000

typedef float v2f __attribute__((ext_vector_type(2)));
typedef float v8f __attribute__((ext_vector_type(8)));

// ---------------------------------------------------------------------------
// Degree / normalization kernels
// ---------------------------------------------------------------------------
__global__ void init_deg_kernel(float* __restrict__ deg, int n) {
    int i = blockIdx.x * blockDim.x + threadIdx.x;
    if (i < n) deg[i] = 1.0f;  // self-loop contributes 1
}

__global__ void count_deg_kernel(const int* __restrict__ dst,
                                 float* __restrict__ deg, int e) {
    int i = blockIdx.x * blockDim.x + threadIdx.x;
    if (i < e) {
        int d = dst[i];
        __hip_atomic_fetch_add(&deg[d], 1.0f, __ATOMIC_RELAXED,
                               __HIP_MEMORY_SCOPE_AGENT);
    }
}

__global__ void rsqrt_kernel(float* __restrict__ deg, int n) {
    int i = blockIdx.x * blockDim.x + threadIdx.x;
    if (i < n) deg[i] = __frsqrt_rn(deg[i]);  // deg >= 1, safe
}

// ---------------------------------------------------------------------------
// Dense GEMM: Y[M,256] = X[M,256] @ W[256,256], fp32 WMMA (16x16x4)
// One wave computes one 16x16 tile. Block = 8 waves covering 8 N-tiles at
// the same M-tile (A-operand reuse through L1). grid = (M/16, 256/(16*8)).
//
// Fragment layouts per CDNA5 ISA 7.12.2:
//  A (16x4 f32): lanes 0-15 -> K = {k0, k0+1}; lanes 16-31 -> K = {k0+2, k0+3};
//                row M = lane & 15
//  B (4x16 f32): mirrored (K rows striped the same way, N = lane & 15)
//  C/D (16x16 f32): vgpr r holds M = r + (lane>=16)*8, N = lane & 15
// ---------------------------------------------------------------------------
__global__ void gemm_wmma_kernel(const float* __restrict__ X,
                                 const float* __restrict__ W,
                                 float* __restrict__ Y, int M) {
    const int lane = threadIdx.x & 31;
    const int wave = threadIdx.x >> 5;
    const int m0 = blockIdx.x * 16;
    const int n0 = (blockIdx.y * 8 + wave) * 16;
    if (m0 >= M) return;

    const int row   = lane & 15;
    const int khalf = (lane >> 4) * 2;  // 0 for lanes 0-15, 2 for lanes 16-31

    const float* __restrict__ xrow = X + (size_t)(m0 + row) * DIM + khalf;
    const float* __restrict__ wcol = W + (size_t)khalf * DIM + (n0 + row);

    v8f acc = {};
#pragma unroll 8
    for (int k0 = 0; k0 < DIM; k0 += 4) {
        v2f a;
        a.x = xrow[k0 + 0];
        a.y = xrow[k0 + 1];
        v2f b;
        b.x = wcol[(size_t)(k0 + 0) * DIM];
        b.y = wcol[(size_t)(k0 + 1) * DIM];
        acc = __builtin_amdgcn_wmma_f32_16x16x4_f32(
            /*neg_a=*/false, a, /*neg_b=*/false, b,
            /*c_mod=*/(short)0, acc, /*reuse_a=*/false, /*reuse_b=*/false);
    }

    float* __restrict__ yout =
        Y + (size_t)(m0 + (lane >> 4) * 8) * DIM + n0 + (lane & 15);
#pragma unroll
    for (int r = 0; r < 8; ++r) yout[(size_t)r * DIM] = acc[r];
}

// ---------------------------------------------------------------------------
// acc[i][:] = dinv[i]^2 * h[i][:]   (self-loop term; fully initializes acc)
// One thread per float4: N*64 threads.
// ---------------------------------------------------------------------------
__global__ void selfloop_init_kernel(const float* __restrict__ h,
                                     const float* __restrict__ dinv,
                                     float* __restrict__ acc, int n) {
    int idx = blockIdx.x * blockDim.x + threadIdx.x;
    int total = n * (DIM / 4);
    if (idx >= total) return;
    int i  = idx >> 6;        // node
    int c4 = idx & 63;        // float4 index within row
    float s = dinv[i];
    s *= s;
    const float4* hp = (const float4*)(h + (size_t)i * DIM) + c4;
    float4 v = *hp;
    v.x *= s; v.y *= s; v.z *= s; v.w *= s;
    ((float4*)(acc + (size_t)i * DIM))[c4] = v;
}

// ---------------------------------------------------------------------------
// Edge scatter: one wave per edge; lane handles 8 channels.
// acc[dst] += dinv[src]*dinv[dst] * h[src]
// ---------------------------------------------------------------------------
__global__ void edge_scatter_kernel(const int* __restrict__ src,
                                    const int* __restrict__ dst,
                                    const float* __restrict__ dinv,
                                    const float* __restrict__ h,
                                    float* __restrict__ acc, int e) {
    const int lane   = threadIdx.x & 31;
    const int gwave  = (blockIdx.x * blockDim.x + threadIdx.x) >> 5;
    const int nwaves = (gridDim.x * blockDim.x) >> 5;

    for (int ei = gwave; ei < e; ei += nwaves) {
        int s = src[ei];
        int d = dst[ei];
        float w = dinv[s] * dinv[d];

        const float4* hp = (const float4*)(h + (size_t)s * DIM) + lane * 2;
        float4 v0 = hp[0];
        float4 v1 = hp[1];

        // speculative prefetch of the next gathered row (global_prefetch_b8)
        int en = ei + nwaves;
        if (en < e) {
            int sn = src[en];
            __builtin_prefetch(h + (size_t)sn * DIM + lane * 8, 0, 3);
        }

        float* ap = acc + (size_t)d * DIM + lane * 8;
        __hip_atomic_fetch_add(ap + 0, w * v0.x, __ATOMIC_RELAXED, __HIP_MEMORY_SCOPE_AGENT);
        __hip_atomic_fetch_add(ap + 1, w * v0.y, __ATOMIC_RELAXED, __HIP_MEMORY_SCOPE_AGENT);
        __hip_atomic_fetch_add(ap + 2, w * v0.z, __ATOMIC_RELAXED, __HIP_MEMORY_SCOPE_AGENT);
        __hip_atomic_fetch_add(ap + 3, w * v0.w, __ATOMIC_RELAXED, __HIP_MEMORY_SCOPE_AGENT);
        __hip_atomic_fetch_add(ap + 4, w * v1.x, __ATOMIC_RELAXED, __HIP_MEMORY_SCOPE_AGENT);
        __hip_atomic_fetch_add(ap + 5, w * v1.y, __ATOMIC_RELAXED, __HIP_MEMORY_SCOPE_AGENT);
        __hip_atomic_fetch_add(ap + 6, w * v1.z, __ATOMIC_RELAXED, __HIP_MEMORY_SCOPE_AGENT);
        __hip_atomic_fetch_add(ap + 7, w * v1.w, __ATOMIC_RELAXED, __HIP_MEMORY_SCOPE_AGENT);
    }
}

// ---------------------------------------------------------------------------
// acc = relu(acc + bias), in place. One thread per float4.
// ---------------------------------------------------------------------------
__global__ void bias_relu_kernel(float* __restrict__ acc,
                                 const float* __restrict__ bias, int n) {
    int idx = blockIdx.x * blockDim.x + threadIdx.x;
    int total = n * (DIM / 4);
    if (idx >= total) return;
    int i  = idx >> 6;
    int c4 = idx & 63;
    float4 b = ((const float4*)bias)[c4];
    float4* p = (float4*)(acc + (size_t)i * DIM) + c4;
    float4 v = *p;
    v.x = fmaxf(v.x + b.x, 0.0f);
    v.y = fmaxf(v.y + b.y, 0.0f);
    v.z = fmaxf(v.z + b.z, 0.0f);
    v.w = fmaxf(v.w + b.w, 0.0f);
    *p = v;
}

// ---------------------------------------------------------------------------
// Heads: out[node] = h[node] . Wa + ba ; out[N + node] = h[node] . Wc + bc
// One wave per node; butterfly reduction over 32 lanes (wave32).
// ---------------------------------------------------------------------------
__global__ void heads_kernel(const float* __restrict__ h,
                             const float* __restrict__ Wa,
                             const float* __restrict__ ba,
                             const float* __restrict__ Wc,
                             const float* __restrict__ bc,
                             float* __restrict__ out, int n) {
    const int lane = threadIdx.x & 31;
    const int node = (blockIdx.x * blockDim.x + threadIdx.x) >> 5;
    if (node >= n) return;

    const float4* hp = (const float4*)(h + (size_t)node * DIM) + lane * 2;
    const float4* wa = (const float4*)Wa + lane * 2;
    const float4* wc = (const float4*)Wc + lane * 2;
    float4 h0 = hp[0], h1 = hp[1];
    float4 a0 = wa[0], a1 = wa[1];
    float4 c0 = wc[0], c1 = wc[1];

    float sa = h0.x * a0.x + h0.y * a0.y + h0.z * a0.z + h0.w * a0.w +
               h1.x * a1.x + h1.y * a1.y + h1.z * a1.z + h1.w * a1.w;
    float sc = h0.x * c0.x + h0.y * c0.y + h0.z * c0.z + h0.w * c0.w +
               h1.x * c1.x + h1.y * c1.y + h1.z * c1.z + h1.w * c1.w;

#pragma unroll
    for (int m = 16; m > 0; m >>= 1) {
        sa += __shfl_xor(sa, m, 32);
        sc += __shfl_xor(sc, m, 32);
    }
    if (lane == 0) {
        out[node]     = sa + ba[0];
        out[n + node] = sc + bc[0];
    }
}

// ---------------------------------------------------------------------------
// Host-side orchestration
// ---------------------------------------------------------------------------
extern "C" void kernel_launch(void* const* d_in, const int* in_sizes, int n_in,
                              void* d_out, int out_size, void* d_ws, size_t ws_size,
                              hipStream_t stream) {
    (void)in_sizes; (void)n_in; (void)out_size; (void)ws_size;

    const float* x   = (const float*)d_in[0];
    const int*   ei  = (const int*)d_in[1];   // [2, E] int32
    const float* W1  = (const float*)d_in[2];
    const float* b1  = (const float*)d_in[3];
    const float* W2  = (const float*)d_in[4];
    const float* b2  = (const float*)d_in[5];
    const float* Wa  = (const float*)d_in[6];
    const float* ba  = (const float*)d_in[7];
    const float* Wc  = (const float*)d_in[8];
    const float* bc  = (const float*)d_in[9];
    float* out = (float*)d_out;

    const int* src = ei;
    const int* dst = ei + N_EDGES;

    // Workspace layout (256B aligned slices)
    char* ws = (char*)d_ws;
    size_t off = 0;
    float* dinv = (float*)(ws + off);
    off += ((size_t)N_NODES * sizeof(float) + 255) & ~(size_t)255;
    float* bufA = (float*)(ws + off);   // GEMM output h
    off += ((size_t)N_NODES * DIM * sizeof(float) + 255) & ~(size_t)255;
    float* bufB = (float*)(ws + off);   // propagation accumulator
    // off += N_NODES*DIM*4  (total ~102.7 MB)

    const int TPB = 256;
    const int nodeBlocks  = (N_NODES + TPB - 1) / TPB;
    const int edgeBlocks  = (N_EDGES + TPB - 1) / TPB;
    const int vecBlocks   = (N_NODES * (DIM / 4) + TPB - 1) / TPB;   // per-float4
    const int edgeWaveBlk = (N_EDGES * 32 + TPB - 1) / TPB;          // wave/edge
    const int headBlocks  = (N_NODES * 32 + TPB - 1) / TPB;          // wave/node
    const dim3 gemmGrid(N_NODES / 16, DIM / (16 * 8));               // 3125 x 2

    // --- normalization ---
    init_deg_kernel<<<nodeBlocks, TPB, 0, stream>>>(dinv, N_NODES);
    count_deg_kernel<<<edgeBlocks, TPB, 0, stream>>>(dst, dinv, N_EDGES);
    rsqrt_kernel<<<nodeBlocks, TPB, 0, stream>>>(dinv, N_NODES);

    // --- layer 1: h = relu(prop(x @ W1) + b1) ---
    gemm_wmma_kernel<<<gemmGrid, TPB, 0, stream>>>(x, W1, bufA, N_NODES);
    selfloop_init_kernel<<<vecBlocks, TPB, 0, stream>>>(bufA, dinv, bufB, N_NODES);
    edge_scatter_kernel<<<edgeWaveBlk, TPB, 0, stream>>>(src, dst, dinv, bufA, bufB, N_EDGES);
    bias_relu_kernel<<<vecBlocks, TPB, 0, stream>>>(bufB, b1, N_NODES);

    // --- layer 2: h = relu(prop(h @ W2) + b2) ---
    gemm_wmma_kernel<<<gemmGrid, TPB, 0, stream>>>(bufB, W2, bufA, N_NODES);
    selfloop_init_kernel<<<vecBlocks, TPB, 0, stream>>>(bufA, dinv, bufB, N_NODES);
    edge_scatter_kernel<<<edgeWaveBlk, TPB, 0, stream>>>(src, dst, dinv, bufA, bufB, N_EDGES);
    bias_relu_kernel<<<vecBlocks, TPB, 0, stream>>>(bufB, b2, N_NODES);

    // --- heads ---
    heads_kernel<<<headBlocks, TPB, 0, stream>>>(bufB, Wa, ba, Wc, bc, out, N_NODES);
}